// CenterlineDiceLoss_16329465659811
// MI455X (gfx1250) — compile-verified
//
#include <hip/hip_runtime.h>
#include <cstdint>
#include <cmath>

#define TILE 64
#define EH    (TILE + 4)    // 68 rows of e tile + halo 2
#define ESTR  72            // E row stride: aligned 72-float rows [x0-4, x0+68)
#define ECH   18            // 72/4 B128 chunks per E row
#define NH    (TILE + 2)    // 66: e_next region (interior + halo 1)
#define NSTR  68            // Nn row stride (value (ny,nx) at ny*NSTR + nx + 3 -> interior 16B aligned)
#define IMG   1024
#define ROWSH 10
#define NPIX  (8 * 1024 * 1024)

// ---------- CDNA5 async global<->LDS helpers (inline asm, ISA §10.7/§15.18) ----------
__device__ __forceinline__ uint32_t lds_off(const void* p) {
    return (uint32_t)(uintptr_t)p;   // low 32 bits of generic ptr = LDS byte address
}
__device__ __forceinline__ void async_g2l_b128(uint32_t ldsOff, const float* gaddr) {
    asm volatile("global_load_async_to_lds_b128 %0, %1, off"
                 :: "v"(ldsOff), "v"(gaddr) : "memory");
}
__device__ __forceinline__ void async_l2g_b128(float* gaddr, uint32_t ldsOff) {
    asm volatile("global_store_async_from_lds_b128 %0, %1, off"
                 :: "v"(gaddr), "v"(ldsOff) : "memory");
}
__device__ __forceinline__ void wait_async0() {
    asm volatile("s_wait_asynccnt 0" ::: "memory");
}

__device__ __forceinline__ float sigmoidf(float v) {
    return 1.0f / (1.0f + expf(-v));
}

// ---------- fused skeleton iteration (z<8: pred chain, z>=8: target chain) ----------
// e_next = minpool3(e) (+inf pad) ; m = maxpool3(e_next) (−inf pad) == open() term
// d = relu(e − m) ; skel = first ? d : skel + relu(d − skel*d)
__global__ __launch_bounds__(256)
void skel_step_kernel(const float* __restrict__ pred_raw,
                      const float* __restrict__ target,
                      const float* __restrict__ e_in,   // 16 planes (steps >= 1)
                      float* __restrict__ e_out,        // 16 planes
                      float* __restrict__ skel,         // 16 planes
                      int first)
{
    __shared__ __align__(16) float E[EH * ESTR];        // 68 x 72   19.1 KB
    __shared__ __align__(16) float H[EH * NH];          // 68 x 66 row-min; reused as 66x64 row-max
    __shared__ __align__(16) float Nn[NH * NSTR + 4];   // 66 x (68+pad)  ~17.6 KB

    const int tid = threadIdx.x;
    const int x0 = blockIdx.x * TILE;
    const int y0 = blockIdx.y * TILE;
    const int z  = blockIdx.z;                 // 0..15
    const bool isPred = (z < 8);
    const size_t planeZ = (size_t)z << 20;

    const float* src;
    size_t sbase;
    if (first) {
        src   = isPred ? pred_raw : target;
        sbase = (size_t)(isPred ? z : z - 8) << 20;
    } else {
        src   = e_in;
        sbase = planeZ;
    }

    // ---- S0: async-stage e tile rows [x0-4, x0+68) x [y0-2, y0+66) via B128 ----
    // Every 4-col chunk is fully in-image or fully out (chunk starts = mult of 4).
    for (int i = tid; i < EH * ECH; i += 256) {
        int ly = i / ECH, j = i - ly * ECH;
        int gy  = y0 - 2 + ly;
        int cy  = min(max(gy, 0), IMG - 1);        // row clamp (fixed up below)
        int gx0 = x0 - 4 + 4 * j;
        if ((unsigned)gx0 <= (unsigned)(IMG - 4)) { // fully in-image chunk
            async_g2l_b128(lds_off(&E[ly * ESTR + 4 * j]),
                           src + sbase + ((size_t)cy << ROWSH) + (size_t)gx0);
        }
        // fully-out chunks skipped: their used cells are +inf-fixed, pad cells never read
    }
    wait_async0();
    __syncthreads();

    // ---- S1: sigmoid (step0, pred half) and/or border fix-up (+inf outside image) ----
    {
        const bool doSig  = (first != 0) && isPred;
        const bool border = (x0 == 0) || (y0 == 0) || (x0 == IMG - TILE) || (y0 == IMG - TILE);
        if (doSig || border) {
            for (int i = tid; i < EH * EH; i += 256) {
                int uy = i / EH, ux = i - uy * EH;      // used window coords
                int gy = y0 - 2 + uy, gx = x0 - 2 + ux;
                int idx = uy * ESTR + ux + 2;
                float v = E[idx];
                if (doSig) v = sigmoidf(v);
                if ((unsigned)gy >= (unsigned)IMG || (unsigned)gx >= (unsigned)IMG)
                    v = __builtin_inff();
                E[idx] = v;
            }
        }
    }
    __syncthreads();

    // ---- S2: row-min3: H[y][j] = min(Eused[y][j..j+2]), 68 x 66 ----
    for (int i = tid; i < EH * NH; i += 256) {
        int y = i / NH, j = i - y * NH;
        const float* r = &E[y * ESTR + j + 2];
        H[i] = fminf(fminf(r[0], r[1]), r[2]);
    }
    __syncthreads();

    // ---- S3: col-min3 -> Nn (66 x 66, stored at +3 col pad); −inf outside image ----
    for (int i = tid; i < NH * NH; i += 256) {
        int ny = i / NH, nx = i - ny * NH;
        int gy = y0 - 1 + ny, gx = x0 - 1 + nx;
        float v;
        if ((unsigned)gy >= (unsigned)IMG || (unsigned)gx >= (unsigned)IMG) {
            v = -__builtin_inff();
        } else {
            const float* c = &H[ny * NH + nx];
            v = fminf(fminf(c[0], c[NH]), c[2 * NH]);
        }
        Nn[ny * NSTR + nx + 3] = v;
    }
    __syncthreads();

    // ---- S4: row-max3 of Nn -> G (66 x 64), reusing H's LDS ----
    float* G = H;
    for (int i = tid; i < NH * TILE; i += 256) {
        int y = i / TILE, j = i - y * TILE;
        const float* r = &Nn[y * NSTR + j + 3];
        G[i] = fmaxf(fmaxf(r[0], r[1]), r[2]);
    }
    __syncthreads();

    // ---- S5: col-max3 -> m; d = relu(e−m); skel (float4); e_next async B128 ----
    for (int i = tid; i < TILE * (TILE / 4); i += 256) {   // 1024 4-px chunks
        int ty  = i / (TILE / 4);
        int tx4 = (i - ty * (TILE / 4)) * 4;

        float4 dv;
        {
            const float* c = &G[ty * TILE + tx4];
            const float* e = &E[(ty + 2) * ESTR + tx4 + 4];
            float m0 = fmaxf(fmaxf(c[0], c[TILE]), c[2 * TILE]);
            float m1 = fmaxf(fmaxf(c[1], c[1 + TILE]), c[1 + 2 * TILE]);
            float m2 = fmaxf(fmaxf(c[2], c[2 + TILE]), c[2 + 2 * TILE]);
            float m3 = fmaxf(fmaxf(c[3], c[3 + TILE]), c[3 + 2 * TILE]);
            dv.x = fmaxf(e[0] - m0, 0.0f);
            dv.y = fmaxf(e[1] - m1, 0.0f);
            dv.z = fmaxf(e[2] - m2, 0.0f);
            dv.w = fmaxf(e[3] - m3, 0.0f);
        }

        size_t g = planeZ + ((size_t)(y0 + ty) << ROWSH) + (size_t)(x0 + tx4);
        float4 s;
        if (first) {
            s = dv;
        } else {
            float4 so = *(const float4*)&skel[g];
            s.x = so.x + fmaxf(dv.x - so.x * dv.x, 0.0f);
            s.y = so.y + fmaxf(dv.y - so.y * dv.y, 0.0f);
            s.z = so.z + fmaxf(dv.z - so.z * dv.z, 0.0f);
            s.w = so.w + fmaxf(dv.w - so.w * dv.w, 0.0f);
        }
        *(float4*)&skel[g] = s;

        // e_next interior chunk straight from LDS (16B-aligned source & dest)
        async_l2g_b128(e_out + g, lds_off(&Nn[(ty + 1) * NSTR + tx4 + 4]));
    }
    wait_async0();
}

// ---------- reductions (deterministic two-stage); sigmoid recomputed inline ----------
__global__ __launch_bounds__(256)
void reduce4_kernel(const float* __restrict__ skel_pred, const float* __restrict__ skel_tgt,
                    const float* __restrict__ target, const float* __restrict__ pred_raw,
                    float* __restrict__ partial, int n4)   // n4 = NPIX/4
{
    float a = 0.f, b = 0.f, c = 0.f, d = 0.f;
    int stride = gridDim.x * 256;
    for (int i = blockIdx.x * 256 + threadIdx.x; i < n4; i += stride) {
        float4 sp = ((const float4*)skel_pred)[i];
        float4 st = ((const float4*)skel_tgt)[i];
        float4 tg = ((const float4*)target)[i];
        float4 pr = ((const float4*)pred_raw)[i];
        a += sp.x * tg.x + sp.y * tg.y + sp.z * tg.z + sp.w * tg.w;
        b += sp.x + sp.y + sp.z + sp.w;
        c += st.x * sigmoidf(pr.x) + st.y * sigmoidf(pr.y)
           + st.z * sigmoidf(pr.z) + st.w * sigmoidf(pr.w);
        d += st.x + st.y + st.z + st.w;
    }
    __shared__ float r[4][256];
    int t = threadIdx.x;
    r[0][t] = a; r[1][t] = b; r[2][t] = c; r[3][t] = d;
    __syncthreads();
    for (int s = 128; s > 0; s >>= 1) {
        if (t < s) {
            r[0][t] += r[0][t + s];
            r[1][t] += r[1][t + s];
            r[2][t] += r[2][t + s];
            r[3][t] += r[3][t + s];
        }
        __syncthreads();
    }
    if (t == 0) {
        partial[blockIdx.x * 4 + 0] = r[0][0];
        partial[blockIdx.x * 4 + 1] = r[1][0];
        partial[blockIdx.x * 4 + 2] = r[2][0];
        partial[blockIdx.x * 4 + 3] = r[3][0];
    }
}

__global__ __launch_bounds__(256)
void final_kernel(const float* __restrict__ partial, int nb, float* __restrict__ out)
{
    float a = 0.f, b = 0.f, c = 0.f, d = 0.f;
    for (int i = threadIdx.x; i < nb; i += 256) {
        a += partial[i * 4 + 0];
        b += partial[i * 4 + 1];
        c += partial[i * 4 + 2];
        d += partial[i * 4 + 3];
    }
    __shared__ float r[4][256];
    int t = threadIdx.x;
    r[0][t] = a; r[1][t] = b; r[2][t] = c; r[3][t] = d;
    __syncthreads();
    for (int s = 128; s > 0; s >>= 1) {
        if (t < s) {
            r[0][t] += r[0][t + s];
            r[1][t] += r[1][t + s];
            r[2][t] += r[2][t + s];
            r[3][t] += r[3][t + s];
        }
        __syncthreads();
    }
    if (t == 0) {
        float tprec = (r[0][0] + 1.0f) / (r[1][0] + 1.0f);
        float tsens = (r[2][0] + 1.0f) / (r[3][0] + 1.0f);
        float cl = 2.0f * tprec * tsens / (tprec + tsens + 1e-7f);
        out[0] = 1.0f - cl;
    }
}

extern "C" void kernel_launch(void* const* d_in, const int* in_sizes, int n_in,
                              void* d_out, int out_size, void* d_ws, size_t ws_size,
                              hipStream_t stream)
{
    (void)in_sizes; (void)n_in; (void)out_size; (void)ws_size;

    const float* pred   = (const float*)d_in[0];
    const float* target = (const float*)d_in[1];
    float* out = (float*)d_out;

    const int N = NPIX;
    float* eA      = (float*)d_ws;       // 2N: 16 planes (pred 0..7, target 8..15)
    float* eB      = eA + 2 * (size_t)N;
    float* skel    = eB + 2 * (size_t)N; // 2N: both skeletons
    float* partial = skel + 2 * (size_t)N;

    dim3 blk(256);
    dim3 grid(IMG / TILE, IMG / TILE, 16);

    // step 0: raw pred (sigmoid fused) and target
    skel_step_kernel<<<grid, blk, 0, stream>>>(pred, target, eA /*unused*/, eA, skel, 1);

    // steps 1..10: ping-pong e
    {
        const float* ein = eA;
        float* eout = eB;
        for (int k = 1; k < 11; ++k) {
            skel_step_kernel<<<grid, blk, 0, stream>>>(pred, target, ein, eout, skel, 0);
            const float* t = eout;
            eout = (float*)ein;
            ein  = t;
        }
    }

    const int NB = 1024;
    reduce4_kernel<<<NB, blk, 0, stream>>>(skel, skel + N, target, pred, partial, N / 4);
    final_kernel<<<1, blk, 0, stream>>>(partial, NB, out);
}